// APDTFlow_33732673142966
// MI455X (gfx1250) — compile-verified
//
#include <hip/hip_runtime.h>
#include <hip/hip_bf16.h>
#include <math.h>

#define S_ 4
#define B_ 256
#define T_ 256
#define H_ 128
#define DW_ 64
#define FH_ 48
#define KA_ 132   // H + 2 (x_t, t) padded to multiple of 4
#define N2_ 256   // 2*H

typedef __attribute__((ext_vector_type(2))) float v2f;
typedef __attribute__((ext_vector_type(8))) float v8f;

__device__ __forceinline__ v8f vzero8() {
  v8f z = {0.f,0.f,0.f,0.f,0.f,0.f,0.f,0.f};
  return z;
}

// D = A(16x4) * B(4x16) + C   (fp32 tensor op, wave32)
__device__ __forceinline__ v8f wmma4(v2f a, v2f b, v8f c) {
  return __builtin_amdgcn_wmma_f32_16x16x4_f32(
      /*neg_a=*/false, a, /*neg_b=*/false, b,
      /*c_mod=*/(short)0, c, /*reuse_a=*/false, /*reuse_b=*/false);
}

// ---------------------------------------------------------------------------
// Stage 1: decompose = (conv1d K=5 -> BN -> ReLU) x2, per scale.
// One workgroup per scale; BN stats reduced in-block over all B*T elements.
// Final output written transposed: xseq[s][t][b] so the ODE reads time columns.
// ---------------------------------------------------------------------------
__global__ void decompose_kernel(const float* __restrict__ x,    // [B][T]
                                 const float* __restrict__ c1w,  // [S][5]
                                 const float* __restrict__ c1b,  // [S]
                                 const float* __restrict__ g1,   // [S]
                                 const float* __restrict__ bb1,  // [S]
                                 const float* __restrict__ c2w,
                                 const float* __restrict__ c2b,
                                 const float* __restrict__ g2,
                                 const float* __restrict__ bb2,
                                 float* __restrict__ h1,         // [S][B][T] scratch
                                 float* __restrict__ xseq)       // [S][T][B]
{
  const int s = blockIdx.x;
  const int tid = threadIdx.x;           // 256 threads
  const int BT = B_ * T_;
  __shared__ float redA[256];
  __shared__ float redB[256];

  float w[5];
  #pragma unroll
  for (int k = 0; k < 5; ++k) w[k] = c1w[s*5 + k];
  float bias = c1b[s];

  float sum = 0.f, sq = 0.f;
  for (int e = tid; e < BT; e += 256) {
    int b = e >> 8, t = e & 255;
    float acc = bias;
    #pragma unroll
    for (int k = 0; k < 5; ++k) {
      int tt = t + k - 2;
      if (tt >= 0 && tt < T_) acc += x[b*T_ + tt] * w[k];
    }
    h1[s*BT + e] = acc;
    sum += acc; sq += acc*acc;
  }
  redA[tid] = sum; redB[tid] = sq;
  __syncthreads();
  for (int off = 128; off > 0; off >>= 1) {
    if (tid < off) { redA[tid] += redA[tid+off]; redB[tid] += redB[tid+off]; }
    __syncthreads();
  }
  float m  = redA[0] / (float)BT;
  float vr = redB[0] / (float)BT - m*m;
  float sc = g1[s] * rsqrtf(vr + 1e-5f);
  float sh = bb1[s];
  __syncthreads();
  for (int e = tid; e < BT; e += 256) {          // own elements only
    float v = h1[s*BT + e];
    v = sc * (v - m) + sh;
    h1[s*BT + e] = v > 0.f ? v : 0.f;
  }
  __threadfence();
  __syncthreads();

  // second conv over normalized h1 (needs neighbors -> after fence)
  #pragma unroll
  for (int k = 0; k < 5; ++k) w[k] = c2w[s*5 + k];
  bias = c2b[s];
  sum = 0.f; sq = 0.f;
  for (int e = tid; e < BT; e += 256) {
    int b = e >> 8, t = e & 255;
    float acc = bias;
    #pragma unroll
    for (int k = 0; k < 5; ++k) {
      int tt = t + k - 2;
      if (tt >= 0 && tt < T_) acc += h1[s*BT + b*T_ + tt] * w[k];
    }
    xseq[s*BT + t*B_ + b] = acc;                  // transposed raw
    sum += acc; sq += acc*acc;
  }
  __syncthreads();
  redA[tid] = sum; redB[tid] = sq;
  __syncthreads();
  for (int off = 128; off > 0; off >>= 1) {
    if (tid < off) { redA[tid] += redA[tid+off]; redB[tid] += redB[tid+off]; }
    __syncthreads();
  }
  m  = redA[0] / (float)BT;
  vr = redB[0] / (float)BT - m*m;
  sc = g2[s] * rsqrtf(vr + 1e-5f);
  sh = bb2[s];
  for (int e = tid; e < BT; e += 256) {          // own transposed elements
    int b = e >> 8, t = e & 255;
    float v = xseq[s*BT + t*B_ + b];
    v = sc * (v - m) + sh;
    xseq[s*BT + t*B_ + b] = v > 0.f ? v : 0.f;
  }
}

// ---------------------------------------------------------------------------
// Stage 2: RK4 Neural ODE. grid = S * (B/16) = 64 workgroups, 128 threads
// (4 waves). Each workgroup integrates a 16-row batch slab for 255 steps.
// GEMM1: (16 x 132)(132 x 64), one 16x16 N-tile per wave, WMMA f32 16x16x4.
// GEMM2: (16 x 64)(64 x 256), four 16x16 N-tiles per wave.
// Waves 0-1 own mean dynamics (update h in LDS); waves 2-3 own logvar
// dynamics (accumulate lv in registers across all steps).
// ---------------------------------------------------------------------------
__global__ void ode_rk4_kernel(const float* __restrict__ xseq,  // [S][T][B]
                               const float* __restrict__ dw1,   // [S][130][64]
                               const float* __restrict__ db1,   // [S][64]
                               const float* __restrict__ dw2,   // [S][64][256]
                               const float* __restrict__ db2,   // [S][256]
                               float* __restrict__ mean_out,    // [S][B][H]
                               float* __restrict__ lv_out)      // [S][B][H]
{
  extern __shared__ float smem[];
  float* w1s = smem;                         // KA_*64   = 8448
  float* w2s = w1s + KA_*64;                 // 64*256   = 16384
  float* b1s = w2s + 64*256;                 // 64
  float* b2s = b1s + 64;                     // 256
  float* A0  = b2s + 256;                    // 16*KA_   = 2112  ([h | x | t | 0 0])
  float* A1  = A0 + 16*KA_;                  // 16*KA_   (probe states)
  float* U   = A1 + 16*KA_;                  // 16*64    (relu(gemm1))

  const int s    = blockIdx.x >> 4;
  const int row0 = (blockIdx.x & 15) * 16;
  const int tid  = threadIdx.x;              // 0..127
  const int wave = tid >> 5;                 // 0..3
  const int lane = tid & 31;
  const int nl   = lane & 15;
  const int hi   = lane >> 4;                // lane-half selects K pair / row+8

  // cooperative weight load to LDS
  for (int i = tid; i < 130*64; i += 128) w1s[i] = dw1[s*130*64 + i];
  for (int i = tid; i < 2*64;  i += 128) w1s[130*64 + i] = 0.f;   // K pad rows
  for (int i = tid; i < 64*256; i += 128) w2s[i] = dw2[s*64*256 + i];
  if (tid < 64)  b1s[tid] = db1[s*64 + tid];
  for (int i = tid; i < 256; i += 128) b2s[i] = db2[s*256 + i];
  for (int i = tid; i < 16*KA_; i += 128) { A0[i] = 0.f; A1[i] = 0.f; }
  __syncthreads();
  if (tid < 16) {
    A0[tid*KA_ + 128] = xseq[(s*T_ + 0)*B_ + row0 + tid];  // x column, idx 0
    A0[tid*KA_ + 129] = 0.f;                               // t column
  }

  const float dt = 1.0f / (float)(T_ - 1);
  v8f ks[4];                 // per-step weighted RK sums (this wave's 4 tiles)
  v8f lv[4];                 // logvar accumulator (waves 2,3)
  #pragma unroll
  for (int i = 0; i < 4; ++i) lv[i] = vzero8();

  for (int step = 0; step < T_ - 1; ++step) {
    const float t0 = step * dt;
    const float t1 = (step + 1) * dt;
    #pragma unroll
    for (int i = 0; i < 4; ++i) ks[i] = vzero8();

    for (int st = 0; st < 4; ++st) {
      const float* A = (st == 0) ? A0 : A1;
      __syncthreads();                               // A buffer ready

      // ---- GEMM1: one 16x16 tile, N range [16*wave, 16*wave+16) ----
      v8f c1 = vzero8();
      const int n0a = wave * 16;
      for (int kk = 0; kk < KA_/4; ++kk) {
        const int kb = 4*kk + 2*hi;
        v2f a, b;
        a.x = A[nl*KA_ + kb];      a.y = A[nl*KA_ + kb + 1];
        b.x = w1s[kb*64 + n0a+nl]; b.y = w1s[(kb+1)*64 + n0a+nl];
        c1 = wmma4(a, b, c1);
      }
      const float bias1 = b1s[n0a + nl];
      #pragma unroll
      for (int v = 0; v < 8; ++v) {
        float uu = c1[v] + bias1;
        uu = uu > 0.f ? uu : 0.f;
        U[(v + (hi << 3))*64 + n0a + nl] = uu;       // relu -> U tile
      }
      __syncthreads();                               // U ready

      // ---- GEMM2: four 16x16 tiles, N range [64*wave, 64*wave+64) ----
      v8f c2[4];
      #pragma unroll
      for (int nt = 0; nt < 4; ++nt) c2[nt] = vzero8();
      for (int kk = 0; kk < 16; ++kk) {
        const int kb = 4*kk + 2*hi;
        v2f a;
        a.x = U[nl*64 + kb]; a.y = U[nl*64 + kb + 1];
        #pragma unroll
        for (int nt = 0; nt < 4; ++nt) {
          const int n0 = wave*64 + nt*16;
          v2f b;
          b.x = w2s[kb*N2_ + n0 + nl]; b.y = w2s[(kb+1)*N2_ + n0 + nl];
          c2[nt] = wmma4(a, b, c2[nt]);
        }
      }

      // ---- RK bookkeeping: kt = stage output; build next probe state ----
      const float wgt  = (st == 1 || st == 2) ? 2.0f : 1.0f;
      const float coef = (st < 2) ? (0.5f * dt) : dt;
      #pragma unroll
      for (int nt = 0; nt < 4; ++nt) {
        const int n0 = wave*64 + nt*16;
        const float bias2 = b2s[n0 + nl];
        #pragma unroll
        for (int v = 0; v < 8; ++v) {
          const float kt = c2[nt][v] + bias2;
          ks[nt][v] += wgt * kt;
          if (wave < 2 && st < 3) {                  // mean-half builds probe h
            const int r = v + (hi << 3);
            const int col = n0 + nl;                 // 0..127
            A1[r*KA_ + col] = A0[r*KA_ + col] + coef * kt;
          }
        }
      }
      if (st < 3 && tid < 16) {                      // probe x, t columns
        const int idx = (st == 2) ? (step + 1) : step;   // argmin tie -> first
        A1[tid*KA_ + 128] = xseq[(s*T_ + idx)*B_ + row0 + tid];
        A1[tid*KA_ + 129] = (st == 2) ? t1 : (t0 + 0.5f*dt);
      }
    }

    __syncthreads();
    const float sixth = dt / 6.0f;
    if (wave < 2) {                                  // h += dt/6 * sum(k)
      #pragma unroll
      for (int nt = 0; nt < 4; ++nt) {
        const int col = wave*64 + nt*16 + nl;
        #pragma unroll
        for (int v = 0; v < 8; ++v)
          A0[(v + (hi << 3))*KA_ + col] += sixth * ks[nt][v];
      }
    } else {                                         // lv += dt/6 * sum(kv)
      #pragma unroll
      for (int nt = 0; nt < 4; ++nt)
        #pragma unroll
        for (int v = 0; v < 8; ++v)
          lv[nt][v] += sixth * ks[nt][v];
    }
    if (tid < 16 && step + 1 < T_ - 1) {             // next step's x, t
      A0[tid*KA_ + 128] = xseq[(s*T_ + step + 1)*B_ + row0 + tid];
      A0[tid*KA_ + 129] = t1;
    }
  }

  __syncthreads();
  for (int i = tid; i < 16*H_; i += 128) {           // means from A0 h-cols
    const int r = i >> 7, c = i & 127;
    mean_out[(s*B_ + row0 + r)*H_ + c] = A0[r*KA_ + c];
  }
  if (wave >= 2) {                                   // logvars from registers
    #pragma unroll
    for (int nt = 0; nt < 4; ++nt) {
      const int col = (wave - 2)*64 + nt*16 + nl;
      #pragma unroll
      for (int v = 0; v < 8; ++v)
        lv_out[(s*B_ + row0 + v + (hi << 3))*H_ + col] = lv[nt][v];
    }
  }
}

// ---------------------------------------------------------------------------
// Stage 3: uncertainty-weighted attention fusion across scales. One block per
// batch row, 128 threads.
// ---------------------------------------------------------------------------
__global__ void attn_fuse_kernel(const float* __restrict__ mean_,  // [S][B][H]
                                 const float* __restrict__ lv_,    // [S][B][H]
                                 const float* __restrict__ a1w,    // [H][32]
                                 const float* __restrict__ a1b,    // [32]
                                 const float* __restrict__ a2w,    // [32]
                                 const float* __restrict__ a2b,    // [1]
                                 float* __restrict__ fused)        // [B][H]
{
  const int b = blockIdx.x;
  const int tid = threadIdx.x;      // 128
  __shared__ float msh[128];
  __shared__ float red[128];
  __shared__ float sc[S_], un[S_], wgt[S_];

  for (int s = 0; s < S_; ++s) {
    msh[tid] = mean_[(s*B_ + b)*H_ + tid];
    red[tid] = __expf(lv_[(s*B_ + b)*H_ + tid]);
    __syncthreads();
    for (int off = 64; off > 0; off >>= 1) {
      if (tid < off) red[tid] += red[tid+off];
      __syncthreads();
    }
    if (tid == 0) un[s] = red[0] / (float)H_;
    __syncthreads();
    float contrib = 0.f;
    if (tid < 32) {
      float acc = a1b[tid];
      for (int h = 0; h < H_; ++h) acc += msh[h] * a1w[h*32 + tid];
      contrib = (acc > 0.f ? acc : 0.f) * a2w[tid];
    }
    red[tid] = contrib;
    __syncthreads();
    for (int off = 64; off > 0; off >>= 1) {
      if (tid < off) red[tid] += red[tid+off];
      __syncthreads();
    }
    if (tid == 0) sc[s] = red[0] + a2b[0];
    __syncthreads();
  }
  if (tid == 0) {
    float z[S_], mx = -1e30f, sum = 0.f;
    for (int s = 0; s < S_; ++s) { z[s] = sc[s] / (un[s] + 1e-6f); mx = fmaxf(mx, z[s]); }
    for (int s = 0; s < S_; ++s) { z[s] = __expf(z[s] - mx); sum += z[s]; }
    for (int s = 0; s < S_; ++s) wgt[s] = z[s] / sum;
  }
  __syncthreads();
  float f = 0.f;
  for (int s = 0; s < S_; ++s) f += wgt[s] * mean_[(s*B_ + b)*H_ + tid];
  fused[b*H_ + tid] = f;
}

// ---------------------------------------------------------------------------
// Stage 4: GRU decoder, 48 sequential steps. One block per batch row,
// 128 threads; thread j owns gate element j of each of the r/z/n thirds.
// ---------------------------------------------------------------------------
__global__ void gru_dec_kernel(const float* __restrict__ fused,  // [B][H]
                               const float* __restrict__ wih,    // [384]
                               const float* __restrict__ whh,    // [384][128]
                               const float* __restrict__ bih,    // [384]
                               const float* __restrict__ bhh,    // [384]
                               const float* __restrict__ fcw,    // [128]
                               const float* __restrict__ fcb,    // [1]
                               const float* __restrict__ lvw,    // [128]
                               const float* __restrict__ lvb,    // [1]
                               const float* __restrict__ tok0,   // [1]
                               float* __restrict__ out)          // [B*FH | B*FH]
{
  const int b = blockIdx.x;
  const int tid = threadIdx.x;     // 128
  __shared__ float hsh[128];
  __shared__ float red[128];
  __shared__ float tok_sh;

  hsh[tid] = fused[b*H_ + tid];
  if (tid == 0) tok_sh = tok0[0];
  __syncthreads();

  for (int step = 0; step < FH_; ++step) {
    const float tok = tok_sh;
    float ghr = bhh[tid], ghz = bhh[tid+128], ghn = bhh[tid+256];
    for (int h = 0; h < H_; ++h) {
      const float hv = hsh[h];
      ghr += hv * whh[ tid       *H_ + h];
      ghz += hv * whh[(tid + 128)*H_ + h];
      ghn += hv * whh[(tid + 256)*H_ + h];
    }
    const float gir = tok*wih[tid]     + bih[tid];
    const float giz = tok*wih[tid+128] + bih[tid+128];
    const float gin = tok*wih[tid+256] + bih[tid+256];
    const float r = 1.f / (1.f + __expf(-(gir + ghr)));
    const float z = 1.f / (1.f + __expf(-(giz + ghz)));
    const float n = tanhf(gin + r*ghn);
    const float hn = (1.f - z)*n + z*hsh[tid];
    __syncthreads();
    hsh[tid] = hn;
    __syncthreads();

    red[tid] = hn * fcw[tid];
    __syncthreads();
    for (int off = 64; off > 0; off >>= 1) {
      if (tid < off) red[tid] += red[tid+off];
      __syncthreads();
    }
    const float pred = red[0] + fcb[0];
    __syncthreads();
    red[tid] = hn * lvw[tid];
    __syncthreads();
    for (int off = 64; off > 0; off >>= 1) {
      if (tid < off) red[tid] += red[tid+off];
      __syncthreads();
    }
    if (tid == 0) {
      out[b*FH_ + step] = pred;
      out[B_*FH_ + b*FH_ + step] = red[0] + lvb[0];
      tok_sh = pred;
    }
    __syncthreads();
  }
}

// ---------------------------------------------------------------------------
extern "C" void kernel_launch(void* const* d_in, const int* in_sizes, int n_in,
                              void* d_out, int out_size, void* d_ws, size_t ws_size,
                              hipStream_t stream)
{
  (void)in_sizes; (void)n_in; (void)out_size; (void)ws_size;

  const float* x    = (const float*)d_in[0];
  /* d_in[1] = t_span: linspace(0,1,T), regenerated analytically in-kernel */
  const float* c1w  = (const float*)d_in[2];
  const float* c1b  = (const float*)d_in[3];
  const float* g1   = (const float*)d_in[4];
  const float* bb1  = (const float*)d_in[5];
  const float* c2w  = (const float*)d_in[6];
  const float* c2b  = (const float*)d_in[7];
  const float* g2   = (const float*)d_in[8];
  const float* bb2  = (const float*)d_in[9];
  const float* dw1  = (const float*)d_in[10];
  const float* db1  = (const float*)d_in[11];
  const float* dw2  = (const float*)d_in[12];
  const float* db2  = (const float*)d_in[13];
  const float* a1w  = (const float*)d_in[14];
  const float* a1b  = (const float*)d_in[15];
  const float* a2w  = (const float*)d_in[16];
  const float* a2b  = (const float*)d_in[17];
  const float* wih  = (const float*)d_in[18];
  const float* whh  = (const float*)d_in[19];
  const float* bih  = (const float*)d_in[20];
  const float* bhh  = (const float*)d_in[21];
  const float* fcw  = (const float*)d_in[22];
  const float* fcb  = (const float*)d_in[23];
  const float* lvw  = (const float*)d_in[24];
  const float* lvb  = (const float*)d_in[25];
  const float* tok0 = (const float*)d_in[26];

  float* ws    = (float*)d_ws;
  float* h1    = ws;                        // S*B*T
  float* xseq  = h1    + S_*B_*T_;          // S*T*B
  float* mean_ = xseq  + S_*B_*T_;          // S*B*H
  float* lv_   = mean_ + S_*B_*H_;          // S*B*H
  float* fused = lv_   + S_*B_*H_;          // B*H

  decompose_kernel<<<S_, 256, 0, stream>>>(x, c1w, c1b, g1, bb1,
                                           c2w, c2b, g2, bb2, h1, xseq);

  const size_t ode_lds = (size_t)(KA_*64 + 64*N2_ + 64 + 256
                                  + 2*16*KA_ + 16*64) * sizeof(float);
  ode_rk4_kernel<<<S_ * (B_/16), 128, ode_lds, stream>>>(xseq, dw1, db1,
                                                         dw2, db2, mean_, lv_);

  attn_fuse_kernel<<<B_, 128, 0, stream>>>(mean_, lv_, a1w, a1b, a2w, a2b, fused);

  gru_dec_kernel<<<B_, 128, 0, stream>>>(fused, wih, whh, bih, bhh,
                                         fcw, fcb, lvw, lvb, tok0, (float*)d_out);
}